// Chamfer_dist_27204322853517
// MI455X (gfx1250) — compile-verified
//
#include <hip/hip_runtime.h>

typedef __attribute__((ext_vector_type(2))) float v2f;
typedef __attribute__((ext_vector_type(8))) float v8f;

#define TILES_PER_WAVE 4
#define WAVES_PER_WG   8
#define ROWS_PER_WG    (WAVES_PER_WG * TILES_PER_WAVE * 16) /* 512 */

// Raw v_min_num_f32: avoids clang's canonicalize (v_max x,x) around fminf.
__device__ __forceinline__ float fmin_raw(float a, float b) {
    float d;
    asm("v_min_num_f32 %0, %1, %2" : "=v"(d) : "v"(a), "v"(b));
    return d;
}

// Pack [x,y,z] -> [x,y,z,x^2+y^2+z^2] for both point clouds.
__global__ __launch_bounds__(256) void pack_kernel(
    const float* __restrict__ a, float4* __restrict__ pa, int na,
    const float* __restrict__ b, float4* __restrict__ pb, int nb)
{
    int i = blockIdx.x * 256 + threadIdx.x;
    if (i < na) {
        float x = a[i * 3 + 0], y = a[i * 3 + 1], z = a[i * 3 + 2];
        pa[i] = make_float4(x, y, z, x * x + y * y + z * z);
    } else if (i < na + nb) {
        int j = i - na;
        float x = b[j * 3 + 0], y = b[j * 3 + 1], z = b[j * 3 + 2];
        pb[j] = make_float4(x, y, z, x * x + y * y + z * z);
    }
}

// For each "query" row, min squared distance to all "ref" points, summed per
// workgroup. d2 tile via V_WMMA_F32_16X16X4_F32:
//   A[m][0..2] = -2*q[m], A[m][3] = 1
//   B[0..2][n] =  r[n],   B[3][n] = ||r[n]||^2
//   C[m][n]    = ||q[m]||^2   (loop-invariant)
// blockIdx.z selects direction: 0 = gts->preds (cham_x), 1 = preds->gts.
template <bool PACKED>
__global__ __launch_bounds__(256) void chamfer_rowmin_kernel(
    const float* __restrict__ gts, const float* __restrict__ preds,
    const float4* __restrict__ gpk, const float4* __restrict__ ppk,
    int N, int M, int nBlkX, int nBlkY,
    float* __restrict__ partX, float* __restrict__ partY)
{
    const int dir  = blockIdx.z;
    const int nBlk = dir ? nBlkY : nBlkX;
    if ((int)blockIdx.x >= nBlk) return;   // uniform per-block exit

    const int Nq = dir ? M : N;
    const int Nr = dir ? N : M;
    const float*  qraw = dir ? preds : gts;
    const float*  rraw = dir ? gts : preds;
    const float4* qpkd = dir ? ppk : gpk;
    const float4* rpkd = dir ? gpk : ppk;
    float* partials = dir ? partY : partX;

    const int lane  = threadIdx.x & 31;
    const int wave  = threadIdx.x >> 5;
    const int batch = blockIdx.y;
    const int half  = lane >> 4;   // 0 = lanes 0-15, 1 = lanes 16-31
    const int l16   = lane & 15;

    const float*  qb  = qraw + (size_t)batch * Nq * 3;
    const float*  rb  = rraw + (size_t)batch * Nr * 3;
    const float4* qb4 = qpkd + (size_t)batch * Nq;
    const float4* rb4 = rpkd + (size_t)batch * Nr;

    v2f a[TILES_PER_WAVE];
    v8f c[TILES_PER_WAVE];
    v8f runmin[TILES_PER_WAVE];
    int tbase[TILES_PER_WAVE];

    #pragma unroll
    for (int t = 0; t < TILES_PER_WAVE; ++t) {
        const int base = blockIdx.x * ROWS_PER_WG + wave * (TILES_PER_WAVE * 16) + t * 16;
        tbase[t] = base;

        // A operand: lanes 0-15 hold (K=0,K=1) of row=lane; lanes 16-31 hold
        // (K=2,K=3) of row=lane-16 (ISA 32-bit A 16x4 layout).
        int rowA = base + (half ? (lane - 16) : lane);
        rowA = rowA < Nq ? rowA : (Nq - 1);
        float q0, q1, q2;
        if (PACKED) {
            float4 vq = qb4[rowA];
            q0 = vq.x; q1 = vq.y; q2 = vq.z;
        } else {
            q0 = qb[rowA * 3 + 0]; q1 = qb[rowA * 3 + 1]; q2 = qb[rowA * 3 + 2];
        }
        a[t].x = half ? (-2.0f * q2) : (-2.0f * q0);
        a[t].y = half ? 1.0f         : (-2.0f * q1);

        // C operand: vgpr v = ||q[base+v]||^2 (lanes 0-15), ||q[base+v+8]||^2
        // (lanes 16-31); distribute via width-16 broadcast shuffles.
        int rowX = base + (half ? (lane - 8) : lane);
        rowX = rowX < Nq ? rowX : (Nq - 1);
        float x2;
        if (PACKED) {
            x2 = qb4[rowX].w;
        } else {
            float p0 = qb[rowX * 3 + 0], p1 = qb[rowX * 3 + 1], p2 = qb[rowX * 3 + 2];
            x2 = p0 * p0 + p1 * p1 + p2 * p2;
        }
        #pragma unroll
        for (int v = 0; v < 8; ++v) c[t][v] = __shfl(x2, v, 16);

        #pragma unroll
        for (int v = 0; v < 8; ++v) runmin[t][v] = 3.4e38f;
    }

    // Load one 16-column ref tile's packed vector for this lane.
    auto loadTile = [&](int j) -> float4 {
        const int col = (j << 4) + l16;
        if (PACKED) {
            return rb4[col];                       // one global_load_b128
        } else {
            float bx = rb[col * 3 + 0], by = rb[col * 3 + 1], bz = rb[col * 3 + 2];
            return make_float4(bx, by, bz, bx * bx + by * by + bz * bz);
        }
    };

    // 4 WMMAs + 32 raw mins per B tile; EXEC stays all-1s throughout.
    auto process = [&](float4 vv) {
        v2f b;
        b.x = half ? vv.z : vv.x;                  // B mirrors A striping
        b.y = half ? vv.w : vv.y;
        #pragma unroll
        for (int t = 0; t < TILES_PER_WAVE; ++t) {
            v8f d = __builtin_amdgcn_wmma_f32_16x16x4_f32(
                false, a[t], false, b, (short)0, c[t], false, false);
            #pragma unroll
            for (int v = 0; v < 8; ++v)
                runmin[t][v] = fmin_raw(runmin[t][v], d[v]);
        }
    };

    // ---- Hot loop: software-pipelined (load j+1 in flight over compute of j)
    const int fullTiles = Nr >> 4;
    if (fullTiles > 0) {
        float4 cur = loadTile(0);
        #pragma unroll 2
        for (int j = 0; j < fullTiles - 1; ++j) {
            float4 nxt = loadTile(j + 1);
            process(cur);
            cur = nxt;
        }
        process(cur);
    }

    // ---- Guarded tail tile (never taken for Nr % 16 == 0)
    if (Nr & 15) {
        const int col = (fullTiles << 4) + l16;
        const bool okc = col < Nr;
        const int  cc  = okc ? col : 0;
        float bx, by, bz, bw;
        if (PACKED) {
            float4 vv = rb4[cc];
            bx = vv.x; by = vv.y; bz = vv.z; bw = vv.w;
        } else {
            bx = rb[cc * 3 + 0]; by = rb[cc * 3 + 1]; bz = rb[cc * 3 + 2];
            bw = bx * bx + by * by + bz * bz;
        }
        bw = okc ? bw : 3.0e38f;
        bx = okc ? bx : 0.0f; by = okc ? by : 0.0f; bz = okc ? bz : 0.0f;
        process(make_float4(bx, by, bz, bw));
    }

    // Per-row min across the 16 columns of each half, clamp >= 0, sum rows.
    float wavesum = 0.0f;
    #pragma unroll
    for (int t = 0; t < TILES_PER_WAVE; ++t) {
        #pragma unroll
        for (int v = 0; v < 8; ++v) {
            float m = fmaxf(runmin[t][v], 0.0f);
            #pragma unroll
            for (int mask = 1; mask <= 8; mask <<= 1)
                m = fmin_raw(m, __shfl_xor(m, mask, 32));
            const int row = tbase[t] + v + (half ? 8 : 0);
            wavesum += (row < Nq) ? m : 0.0f;
        }
    }
    wavesum += __shfl_xor(wavesum, 16, 32);

    __shared__ float lds[WAVES_PER_WG];
    if (lane == 0) lds[wave] = wavesum;
    __syncthreads();
    if (threadIdx.x == 0) {
        float s = 0.0f;
        #pragma unroll
        for (int w = 0; w < WAVES_PER_WG; ++w) s += lds[w];
        partials[batch * nBlk + blockIdx.x] = s;
    }
}

// Deterministic final combine: out = sx * sum(px) + sy * sum(py).
__global__ __launch_bounds__(256) void chamfer_finalize_kernel(
    const float* __restrict__ px, int nx, float sx,
    const float* __restrict__ py, int ny, float sy,
    float* __restrict__ out)
{
    __shared__ float lds[256];
    float acc = 0.0f;
    for (int i = threadIdx.x; i < nx; i += 256) acc += px[i] * sx;
    for (int i = threadIdx.x; i < ny; i += 256) acc += py[i] * sy;
    lds[threadIdx.x] = acc;
    __syncthreads();
    for (int s = 128; s > 0; s >>= 1) {
        if ((int)threadIdx.x < s) lds[threadIdx.x] += lds[threadIdx.x + s];
        __syncthreads();
    }
    if (threadIdx.x == 0) out[0] = lds[0];
}

extern "C" void kernel_launch(void* const* d_in, const int* in_sizes, int n_in,
                              void* d_out, int out_size, void* d_ws, size_t ws_size,
                              hipStream_t stream) {
    const float* gts   = (const float*)d_in[0];
    const float* preds = (const float*)d_in[1];
    const int B = 8, D = 3;
    const int N = in_sizes[0] / (B * D);
    const int M = in_sizes[1] / (B * D);

    const int nBlkX = (N + ROWS_PER_WG - 1) / ROWS_PER_WG;
    const int nBlkY = (M + ROWS_PER_WG - 1) / ROWS_PER_WG;
    const int nPtsQ = B * N, nPtsR = B * M;

    // ws layout: [partX | partY | (16B-aligned) packedG | packedP]
    float* partX = (float*)d_ws;
    float* partY = partX + (size_t)nBlkX * B;
    size_t partBytes = ((size_t)(nBlkX + nBlkY) * B * sizeof(float) + 15) & ~(size_t)15;
    float4* gpk = (float4*)((char*)d_ws + partBytes);
    float4* ppk = gpk + nPtsQ;
    size_t needed = partBytes + ((size_t)nPtsQ + nPtsR) * sizeof(float4);

    const bool usePacked = ws_size >= needed;
    const dim3 grid(nBlkX > nBlkY ? nBlkX : nBlkY, B, 2);

    if (usePacked) {
        int nPack = nPtsQ + nPtsR;
        pack_kernel<<<(nPack + 255) / 256, 256, 0, stream>>>(gts, gpk, nPtsQ,
                                                             preds, ppk, nPtsR);
        chamfer_rowmin_kernel<true><<<grid, 256, 0, stream>>>(
            gts, preds, gpk, ppk, N, M, nBlkX, nBlkY, partX, partY);
    } else {
        chamfer_rowmin_kernel<false><<<grid, 256, 0, stream>>>(
            gts, preds, gpk, ppk, N, M, nBlkX, nBlkY, partX, partY);
    }

    chamfer_finalize_kernel<<<1, 256, 0, stream>>>(
        partX, nBlkX * B, 1.0f / ((float)B * (float)N),
        partY, nBlkY * B, 1.0f / ((float)B * (float)M),
        (float*)d_out);
}